// Abstractinator_55989193671336
// MI455X (gfx1250) — compile-verified
//
#include <hip/hip_runtime.h>
#include <hip/hip_bf16.h>

// Problem constants (match the reference)
#define B_      8
#define S_      8192
#define K_      512            // per-stage codebook
#define DEPTH_  2
#define BS_     (B_ * S_)      // 65536 positions
#define BASE_   (K_ * K_)      // 262144 = K^DEPTH
#define RTOT_   (DEPTH_ * BS_) // 131072 logsumexp rows of length K
#define NB_DIGIT 512           // blocks for digit CE kernel
#define NB_SP    256           // blocks for special CE kernel (256*256 == BS_)

typedef __attribute__((ext_vector_type(2))) float v2f;
typedef __attribute__((ext_vector_type(4))) float v4f;
typedef __attribute__((ext_vector_type(8))) float v8f;

// ---------------------------------------------------------------------------
// Wave32 full-sum reduction via V_WMMA_F32_16X16X4_F32.
// A (16x4 f32): lane L (L<16) supplies A[L][0]=x, A[L][1]=0;
//               lane L+16 supplies A[L][2]=x, A[L][3]=0  (ISA 7.12.2 layout).
// B = ones(4x16)  =>  D[m][n] = acc[m] + acc[m+16]  for every column n.
// C/D layout: lane n, VGPR j holds D[j][n] (n<16) / D[j+8][n-16].
// Sum the 8 D VGPRs per lane, then one shfl_xor(16) -> every lane = total sum.
// Exact in fp32; requires EXEC all-ones (call only from uniform control flow).
// ---------------------------------------------------------------------------
__device__ __forceinline__ float wave_sum_wmma(float x) {
  v2f a; a.x = x;    a.y = 0.0f;
  v2f b; b.x = 1.0f; b.y = 1.0f;
  v8f c = {};
  c = __builtin_amdgcn_wmma_f32_16x16x4_f32(false, a, false, b,
                                            (short)0, c, false, false);
  float q = c[0] + c[1] + c[2] + c[3] + c[4] + c[5] + c[6] + c[7];
  q += __shfl_xor(q, 16, 32);
  return q;
}

// ---------------------------------------------------------------------------
// Kernel 1: length-preserving EOP insertion. One 256-thread block per batch
// row; each thread owns a contiguous chunk of 32 tokens (needed for in-order
// cumsum). Uses cumsum(capped) == min(cumsum(raw), pad_slots).
// ---------------------------------------------------------------------------
__global__ void eop_insert_kernel(const int* __restrict__ seq,
                                  const int* __restrict__ endm,
                                  const int* __restrict__ kpm,
                                  const int* __restrict__ seg,
                                  const int* __restrict__ eop_id_p,
                                  const int* __restrict__ pad_id_p,
                                  float* __restrict__ out) {
  const int b   = blockIdx.x;
  const int tid = threadIdx.x;
  const int base = tid * 32;
  const int row = b * S_;
  __shared__ int sscan[256];
  __shared__ int sred[256];

  // per-thread partials over the 32-token chunk
  int cntEnd = 0, cntPad = 0, mseg = -1;
  for (int j = 0; j < 32; ++j) {
    const int s  = base + j;
    const int em = endm[row + s];
    const int kp = kpm[row + s];
    cntEnd += em;
    cntPad += kp;
    if (!kp) mseg = max(mseg, seg[row + s]);
  }

  // inclusive scan of end-mask counts (Hillis-Steele)
  sscan[tid] = cntEnd;
  __syncthreads();
  for (int off = 1; off < 256; off <<= 1) {
    int v = (tid >= off) ? sscan[tid - off] : 0;
    __syncthreads();
    sscan[tid] += v;
    __syncthreads();
  }
  const int exclBase = sscan[tid] - cntEnd;

  // total pad slots
  sred[tid] = cntPad;
  __syncthreads();
  for (int off = 128; off > 0; off >>= 1) {
    if (tid < off) sred[tid] += sred[tid + off];
    __syncthreads();
  }
  const int pad_slots = sred[0];
  __syncthreads();

  // max seg over valid tokens
  sred[tid] = mseg;
  __syncthreads();
  for (int off = 128; off > 0; off >>= 1) {
    if (tid < off) sred[tid] = max(sred[tid], sred[tid + off]);
    __syncthreads();
  }
  const int last_seg = max(sred[0], 0);

  float* __restrict__ outTok = out + row;             // out (as float)
  float* __restrict__ outKpm = out + BS_ + row;       // out_kpm (0/1 float)
  float* __restrict__ outSeg = out + 2 * BS_ + row;   // out_seg (as float)
  const int pad_id = pad_id_p[0];
  const int eop_id = eop_id_p[0];

  // fill defaults (pads)
  const float padf = (float)pad_id;
  const float lsegf = (float)last_seg;
  for (int j = 0; j < 32; ++j) {
    const int s = base + j;
    outTok[s] = padf;
    outKpm[s] = 1.0f;
    outSeg[s] = lsegf;
  }
  __syncthreads();

  // scatter real tokens + EOPs (targets are provably disjoint)
  int csum = exclBase;
  for (int j = 0; j < 32; ++j) {
    const int s  = base + j;
    const int em = endm[row + s];
    const int kp = kpm[row + s];
    csum += em;
    const int capped = (em && (csum <= pad_slots)) ? 1 : 0;
    const int c = min(csum, pad_slots);
    const int shift = c - capped;
    if (!kp) {
      const int sq = seq[row + s];
      const int sg = seg[row + s];
      const int tgt = s + shift;
      if (tgt < S_) {
        outTok[tgt] = (float)sq;
        outKpm[tgt] = 0.0f;
        outSeg[tgt] = (float)sg;
      }
      if (capped) {
        const int te = tgt + 1;
        if (te < S_) {
          outTok[te] = (float)eop_id;
          outKpm[te] = 0.0f;
          outSeg[te] = (float)sg;
        }
      }
    }
  }
}

// ---------------------------------------------------------------------------
// Kernel 2: stage-wise digit CE. One wave32 per (depth,b,s) row of K=512.
// Coalesced B128 non-temporal loads; butterfly shfl reductions for
// max / sumexp / gathered-logit; per-wave sum reduced with V_WMMA; per-block
// partial written to workspace (deterministic).
// ---------------------------------------------------------------------------
__global__ void digit_ce_kernel(const float* __restrict__ stage,
                                const int* __restrict__ targets,
                                const int* __restrict__ tkpm,
                                float* __restrict__ partials) {
  const int lane   = threadIdx.x & 31;
  const int warp   = threadIdx.x >> 5;
  const int waveId = blockIdx.x * (blockDim.x >> 5) + warp;
  const int nWaves = gridDim.x * (blockDim.x >> 5);

  float acc = 0.0f;
  int it = 0;
  for (int rowi = waveId; rowi < RTOT_; rowi += nWaves, ++it) {
    const int d   = rowi >> 16;          // rowi / BS_
    const int pos = rowi & (BS_ - 1);
    const int t   = targets[pos];
    const int kp  = tkpm[pos];
    const bool valid = (t < BASE_) && (kp == 0);   // wave-uniform
    if (!valid) continue;                          // skip the 2KB row entirely

    const int dig = (d == 0) ? (t & (K_ - 1)) : ((t >> 9) & (K_ - 1));

    const v4f* __restrict__ rp =
        (const v4f*)(stage + (size_t)rowi * (size_t)K_);
    const v4f v0 = __builtin_nontemporal_load(rp + lane);
    const v4f v1 = __builtin_nontemporal_load(rp + 32 + lane);
    const v4f v2 = __builtin_nontemporal_load(rp + 64 + lane);
    const v4f v3 = __builtin_nontemporal_load(rp + 96 + lane);

    // row max
    float m = fmaxf(fmaxf(fmaxf(v0.x, v0.y), fmaxf(v0.z, v0.w)),
                    fmaxf(fmaxf(v1.x, v1.y), fmaxf(v1.z, v1.w)));
    m = fmaxf(m, fmaxf(fmaxf(fmaxf(v2.x, v2.y), fmaxf(v2.z, v2.w)),
                       fmaxf(fmaxf(v3.x, v3.y), fmaxf(v3.z, v3.w))));
    for (int off = 16; off > 0; off >>= 1)
      m = fmaxf(m, __shfl_xor(m, off, 32));

    // gathered logit x[dig]: element index = (i*32+lane)*4 + jj
    const int dig4 = dig >> 2, di = dig4 >> 5, dl = dig4 & 31, dj = dig & 3;
    float xd = 0.0f;
    if (lane == dl) {
      const v4f vv = (di == 0) ? v0 : (di == 1) ? v1 : (di == 2) ? v2 : v3;
      xd = (dj == 0) ? vv.x : (dj == 1) ? vv.y : (dj == 2) ? vv.z : vv.w;
    }

    // sum of exp(x - m)
    float se = __expf(v0.x - m) + __expf(v0.y - m) + __expf(v0.z - m) + __expf(v0.w - m)
             + __expf(v1.x - m) + __expf(v1.y - m) + __expf(v1.z - m) + __expf(v1.w - m)
             + __expf(v2.x - m) + __expf(v2.y - m) + __expf(v2.z - m) + __expf(v2.w - m)
             + __expf(v3.x - m) + __expf(v3.y - m) + __expf(v3.z - m) + __expf(v3.w - m);
    for (int off = 16; off > 0; off >>= 1) {
      se += __shfl_xor(se, off, 32);
      xd += __shfl_xor(xd, off, 32);   // only one lane nonzero -> sum == x[dig]
    }

    const float ce = m + __logf(se) - xd;
    if (lane == (it & 31)) acc += ce;  // rotate accumulating lane
  }

  // wave32 exact sum via WMMA (uniform point, EXEC all-ones)
  const float wsum = wave_sum_wmma(acc);

  __shared__ float sw[8];
  if (lane == 0) sw[warp] = wsum;
  __syncthreads();
  if (threadIdx.x == 0) {
    float s = 0.0f;
    for (int i = 0; i < 8; ++i) s += sw[i];
    partials[blockIdx.x] = s;
  }
}

// ---------------------------------------------------------------------------
// Kernel 3: special-head CE over (B,S,4) + valid-count for digit denom.
// One thread per position; per-block {sp_sum, sp_count, valid_count} partials.
// ---------------------------------------------------------------------------
__global__ void special_ce_kernel(const float* __restrict__ sp_logits,
                                  const int* __restrict__ targets,
                                  const int* __restrict__ tkpm,
                                  float* __restrict__ partials) {
  const int tid = threadIdx.x;
  const int pos = blockIdx.x * blockDim.x + tid;   // exactly BS_ threads

  const int t  = targets[pos];
  const int kp = tkpm[pos];
  const float validf = ((t < BASE_) && (kp == 0)) ? 1.0f : 0.0f;
  const float spmask = ((t >= BASE_) && (kp == 0)) ? 1.0f : 0.0f;
  const int loc = min(max(t - BASE_, 0), 3);

  const v4f x = ((const v4f*)sp_logits)[pos];
  const float m = fmaxf(fmaxf(x.x, x.y), fmaxf(x.z, x.w));
  const float se = __expf(x.x - m) + __expf(x.y - m) +
                   __expf(x.z - m) + __expf(x.w - m);
  const float xl = (loc == 0) ? x.x : (loc == 1) ? x.y : (loc == 2) ? x.z : x.w;
  const float ce = (m + __logf(se) - xl) * spmask;

  __shared__ float s0[256], s1[256], s2[256];
  s0[tid] = ce; s1[tid] = spmask; s2[tid] = validf;
  __syncthreads();
  for (int off = 128; off > 0; off >>= 1) {
    if (tid < off) {
      s0[tid] += s0[tid + off];
      s1[tid] += s1[tid + off];
      s2[tid] += s2[tid + off];
    }
    __syncthreads();
  }
  if (tid == 0) {
    partials[blockIdx.x * 3 + 0] = s0[0];
    partials[blockIdx.x * 3 + 1] = s1[0];
    partials[blockIdx.x * 3 + 2] = s2[0];
  }
}

// ---------------------------------------------------------------------------
// Kernel 4: deterministic finalize (sequential sums; ~1.3k adds).
// ---------------------------------------------------------------------------
__global__ void finalize_kernel(const float* __restrict__ dpart,
                                const float* __restrict__ spart,
                                float* __restrict__ out) {
  if (threadIdx.x == 0 && blockIdx.x == 0) {
    float ds = 0.0f;
    for (int i = 0; i < NB_DIGIT; ++i) ds += dpart[i];
    float ss = 0.0f, sc = 0.0f, vc = 0.0f;
    for (int i = 0; i < NB_SP; ++i) {
      ss += spart[i * 3 + 0];
      sc += spart[i * 3 + 1];
      vc += spart[i * 3 + 2];
    }
    out[3 * BS_ + 0] = ds / fmaxf(vc, 1.0f);   // digit_ce
    out[3 * BS_ + 1] = ss / fmaxf(sc, 1.0f);   // special_ce
  }
}

// ---------------------------------------------------------------------------
// inputs: 0 stage_logits f32 (2,8,8192,512), 1 special_logits f32 (8,8192,4),
//         2 seq i32, 3 end_mask, 4 kpm, 5 seg i32, 6 targets i32, 7 tgt_kpm,
//         8 eop_id, 9 pad_id
// d_out (float): [out BS][out_kpm BS][out_seg BS][digit_ce][special_ce]
// ---------------------------------------------------------------------------
extern "C" void kernel_launch(void* const* d_in, const int* in_sizes, int n_in,
                              void* d_out, int out_size, void* d_ws, size_t ws_size,
                              hipStream_t stream) {
  const float* stage   = (const float*)d_in[0];
  const float* spl     = (const float*)d_in[1];
  const int*   seq     = (const int*)d_in[2];
  const int*   endm    = (const int*)d_in[3];
  const int*   kpm     = (const int*)d_in[4];
  const int*   seg     = (const int*)d_in[5];
  const int*   targets = (const int*)d_in[6];
  const int*   tkpm    = (const int*)d_in[7];
  const int*   eop_id  = (const int*)d_in[8];
  const int*   pad_id  = (const int*)d_in[9];
  float* out = (float*)d_out;

  float* dpart = (float*)d_ws;               // NB_DIGIT floats
  float* spart = dpart + NB_DIGIT;           // NB_SP * 3 floats

  eop_insert_kernel<<<B_, 256, 0, stream>>>(seq, endm, kpm, seg, eop_id, pad_id, out);
  digit_ce_kernel<<<NB_DIGIT, 256, 0, stream>>>(stage, targets, tkpm, dpart);
  special_ce_kernel<<<NB_SP, 256, 0, stream>>>(spl, targets, tkpm, spart);
  finalize_kernel<<<1, 32, 0, stream>>>(dpart, spart, out);
}